// ComplexTVDenoiser_12189117186645
// MI455X (gfx1250) — compile-verified
//
#include <hip/hip_runtime.h>
#include <stdint.h>

// ---------------- problem constants (from reference) ----------------
#define TV_TAU   0.01f
#define TV_SIGMA 12.5f          // 1/TAU/8
#define TV_RHO   1.99f
#define TV_NIT   50
#define TV_H     512
#define TV_W     512
#define TV_NIMG  8
#define TILE     32
#define PAD_W    514            // 512 + zero ring on each side
#define U2_ROW_F (PAD_W * 2)    // floats per padded u2 row = 1028
#define LROW     68             // floats per LDS u2 row: 34 px * 2 ch
#define SROW     64             // floats per LDS u2' staging row: 32 px * 2 ch

typedef unsigned int tdm_v4u __attribute__((ext_vector_type(4)));
typedef int          tdm_v8i __attribute__((ext_vector_type(8)));
typedef int          tdm_v4i __attribute__((ext_vector_type(4)));

// Build the two descriptor groups for a 2D f32 tile (tdim0 x tdim1 elements,
// row stride strideF elements) at global byte address gaddr <-> LDS offset ldsa.
__device__ __forceinline__ void tdm_make_desc(unsigned long long gaddr, unsigned ldsa,
                                              unsigned tdim0, unsigned tdim1,
                                              unsigned strideF, tdm_v4u& g0, tdm_v8i& g1) {
  g0.x = 1u;                                        // count=1 (valid), no gather
  g0.y = ldsa;                                      // lds_addr [63:32]
  g0.z = (unsigned)gaddr;                           // global_addr[31:0]
  g0.w = (unsigned)((gaddr >> 32) & 0x01FFFFFFull)  // global_addr[56:32]
         | (2u << 30);                              // type = 2 ("image")
  g1[0] = (int)(2u << 16);        // data_size = 2 -> 4 bytes; no multicast/iterate/pad
  g1[1] = (int)(tdim0 << 16);     // tensor_dim0[15:0] (tile always in-bounds)
  g1[2] = (int)(tdim1 << 16);     // tensor_dim0 hi = 0; tensor_dim1[15:0]
  g1[3] = (int)(tdim0 << 16);     // tensor_dim1 hi = 0; tile_dim0
  g1[4] = (int)tdim1;             // tile_dim1; tile_dim2 = 0 (2D)
  g1[5] = (int)strideF;           // tensor_dim0_stride (elements)
  g1[6] = 0;                      // stride hi; tensor_dim1_stride unused
  g1[7] = 0;
}

// ---------------- init: x2(=d_out) <- y ; u2a,u2b <- 0 (incl. zero rings) ----
__global__ void __launch_bounds__(256) tv_init(const float* __restrict__ y,
                                               float* __restrict__ x2_init,
                                               float* __restrict__ u2a,
                                               float* __restrict__ u2b) {
  const size_t NU2 = (size_t)TV_NIMG * PAD_W * PAD_W * 2;
  const size_t NX  = (size_t)TV_NIMG * TV_H * TV_W;
  size_t i      = (size_t)blockIdx.x * blockDim.x + threadIdx.x;
  size_t stride = (size_t)gridDim.x * blockDim.x;
  for (size_t k = i; k < NU2; k += stride) { u2a[k] = 0.0f; u2b[k] = 0.0f; }
  for (size_t k = i; k < NX;  k += stride) { x2_init[k] = y[k]; }
}

// ---------------- one fused primal+dual iteration ----------------
// u2 buffers are (8, 514, 514, 2) with a zero ring; x2/y are (8, 512, 512).
__global__ void __launch_bounds__(256) tv_step(const float* __restrict__ x2_in,
                                               const float* __restrict__ u2_in,
                                               const float* __restrict__ y,
                                               const float* __restrict__ ths_p,
                                               float* __restrict__ x2_out,
                                               float* __restrict__ u2_out) {
  __shared__ float su2[34 * LROW];    // old u2 halo tile (rows h0-1..h0+32, padded h0..h0+33)
  __shared__ float sv[33 * 33];       // v = 2x - x2 over tile + bottom/right halo
  __shared__ float sunew[TILE*SROW];  // new u2 tile, staged for TDM store

  const unsigned tid = threadIdx.x;
  const unsigned h0  = blockIdx.y * TILE;
  const unsigned w0  = blockIdx.x * TILE;
  const unsigned img = blockIdx.z;

  // ---- Issue TDM load of the u2 halo tile (wave 0): 34 rows x 68 f32,
  // contiguous per row (34 px * 2 ch), row stride 1028 f32.
  if (tid < 32u) {
    unsigned long long gaddr =
        (unsigned long long)(uintptr_t)u2_in +
        4ull * ((((unsigned long long)img * PAD_W + h0) * PAD_W + w0) * 2ull);
    tdm_v4u g0; tdm_v8i g1;
    tdm_make_desc(gaddr, (unsigned)(uintptr_t)(&su2[0]), 68u, 34u, U2_ROW_F, g0, g1);
    tdm_v4i gz  = {0, 0, 0, 0};
    tdm_v8i gz8 = {0, 0, 0, 0, 0, 0, 0, 0};
    __builtin_amdgcn_tensor_load_to_lds(g0, g1, gz, gz, gz8, 0);
  }

  // ---- Overlap the DMA: all waves prefetch their x2/y elements (L2 hits).
  float x2r[5], yr[5];
  unsigned gidx[5];
#pragma unroll
  for (int k = 0; k < 5; ++k) {
    unsigned i = tid + (unsigned)k * 256u;
    if (i < 33u * 33u) {
      unsigned eh = i / 33u, ew = i - eh * 33u;
      unsigned h = h0 + eh, w = w0 + ew;
      unsigned hc = h < 511u ? h : 511u;    // clamp: out-of-image extended pixels
      unsigned wc = w < 511u ? w : 511u;    // feed only guarded sv entries
      unsigned g = ((img * TV_H + hc) * TV_W + wc);
      gidx[k] = g;
      x2r[k]  = x2_in[g];
      yr[k]   = y[g];
    }
  }
  const float ths = ths_p[0];

  if (tid < 32u) __builtin_amdgcn_s_wait_tensorcnt(0);
  __syncthreads();

  // ---- Phase 1: x-update + relaxation over extended region (33x33).
  // div(h,w) = u0[h-1,w] - u0[h,w] + u1[h,w-1] - u1[h,w]   (branch-free:
  // top/left terms hit the zero ring; bottom/right dual entries are
  // identically zero by induction from the forward-difference padding).
  const float xscale = 1.0f / (1.0f + TV_TAU);
#pragma unroll
  for (int k = 0; k < 5; ++k) {
    unsigned i = tid + (unsigned)k * 256u;
    if (i < 33u * 33u) {
      unsigned eh = i / 33u, ew = i - eh * 33u;
      float x2v = x2r[k];
      float dvg = su2[eh * LROW + (ew + 1u) * 2u]
                - su2[(eh + 1u) * LROW + (ew + 1u) * 2u]
                + su2[(eh + 1u) * LROW + ew * 2u + 1u]
                - su2[(eh + 1u) * LROW + (ew + 1u) * 2u + 1u];
      float xn = (x2v + TV_TAU * (yr[k] - dvg)) * xscale;
      sv[eh * 33u + ew] = 2.0f * xn - x2v;
      if (eh < 32u && ew < 32u)                     // tile pixels only (in-image)
        x2_out[gidx[k]] = x2v + TV_RHO * (xn - x2v);
    }
  }
  __syncthreads();

  // ---- Phase 2: dual update u = prox(u2 + SIGMA * grad(v)), relaxation.
  // Results staged in LDS, then DMA'd out with one tensor_store_from_lds.
  const float inv_ths = 1.0f / ths;
#pragma unroll
  for (int k = 0; k < 4; ++k) {
    unsigned i = tid + (unsigned)k * 256u;
    unsigned th = i >> 5, tw = i & 31u;
    unsigned h = h0 + th, w = w0 + tw;
    float v00 = sv[th * 33u + tw];
    float dh = (h < 511u) ? sv[(th + 1u) * 33u + tw] - v00 : 0.0f;
    float dw = (w < 511u) ? sv[th * 33u + tw + 1u] - v00 : 0.0f;
    float u20 = su2[(th + 1u) * LROW + (tw + 1u) * 2u];
    float u21 = su2[(th + 1u) * LROW + (tw + 1u) * 2u + 1u];
    float un0 = fmaf(TV_SIGMA, dh, u20);
    float un1 = fmaf(TV_SIGMA, dw, u21);
    float n = sqrtf(fmaf(un0, un0, un1 * un1));
    float s = 1.0f / fmaxf(n * inv_ths, 1.0f);
    float r0 = un0 * s, r1 = un1 * s;
    sunew[th * SROW + tw * 2u]      = u20 + TV_RHO * (r0 - u20);
    sunew[th * SROW + tw * 2u + 1u] = u21 + TV_RHO * (r1 - u21);
  }
  __syncthreads();

  // ---- TDM store of the new u2 tile: 32 rows x 64 f32 into the padded
  // interior at (img, h0+1, w0+1). Zero rings are never written.
  if (tid < 32u) {
    unsigned long long gaddr =
        (unsigned long long)(uintptr_t)u2_out +
        4ull * ((((unsigned long long)img * PAD_W + (h0 + 1u)) * PAD_W + (w0 + 1u)) * 2ull);
    tdm_v4u g0; tdm_v8i g1;
    tdm_make_desc(gaddr, (unsigned)(uintptr_t)(&sunew[0]), 64u, 32u, U2_ROW_F, g0, g1);
    tdm_v4i gz  = {0, 0, 0, 0};
    tdm_v8i gz8 = {0, 0, 0, 0, 0, 0, 0, 0};
    __builtin_amdgcn_tensor_store_from_lds(g0, g1, gz, gz, gz8, 0);
    // No explicit wait: S_ENDPGM performs an implicit wait-idle (TENSORcnt),
    // so the store completes before the workgroup is considered finished.
  }
}

// ---------------- host-side orchestration ----------------
extern "C" void kernel_launch(void* const* d_in, const int* in_sizes, int n_in,
                              void* d_out, int out_size, void* d_ws, size_t ws_size,
                              hipStream_t stream) {
  (void)in_sizes; (void)n_in; (void)out_size;

  const float* y     = (const float*)d_in[0];
  const float* ths   = (const float*)d_in[1];
  float*       out   = (float*)d_out;
  char*        ws    = (char*)d_ws;

  const size_t U2B = (size_t)TV_NIMG * PAD_W * PAD_W * 2 * sizeof(float); // 16.9 MB
  const size_t X2B = (size_t)TV_NIMG * TV_H * TV_W * sizeof(float);       // 8.4 MB
  if (ws_size < 2 * U2B + X2B) return;  // need ~42.2 MB of scratch

  float* u2a = (float*)(ws);
  float* u2b = (float*)(ws + U2B);
  float* x2a = (float*)(ws + 2 * U2B);

  // x2 ping-pongs between d_out (initial state = y) and x2a. With 50
  // iterations, even iterations write x2a and odd write d_out, so the
  // final x2 (after it=49) lands in d_out with no special-casing.
  tv_init<<<2048, 256, 0, stream>>>(y, out, u2a, u2b);

  dim3 grid(TV_W / TILE, TV_H / TILE, TV_NIMG);
  const float* x2i = out;  float* x2o = x2a;
  const float* u2i = u2a;  float* u2o = u2b;
  for (int it = 0; it < TV_NIT; ++it) {
    tv_step<<<grid, 256, 0, stream>>>(x2i, u2i, y, ths, x2o, u2o);
    { const float* t = x2i; x2i = x2o; x2o = (float*)t; }
    { const float* t = u2i; u2i = u2o; u2o = (float*)t; }
  }
}